// SheafGCNLayer2_79027398246778
// MI455X (gfx1250) — compile-verified
//
#include <hip/hip_runtime.h>

#define D    64
#define NT   8
#define TILE 16
#define HDR_INTS 64   // [0..7]=counts, [8..16]=starts(9), [17..24]=cursors

typedef float v2f __attribute__((ext_vector_type(2)));
typedef float v8f __attribute__((ext_vector_type(8)));

// ---------------- sort-by-type machinery ----------------

__global__ void init_ws_kernel(int* hdr, int* ssrc, int cap) {
    int i = blockIdx.x * blockDim.x + threadIdx.x;
    if (i < cap) ssrc[i] = -1;          // sentinel: padding rows
    if (i < HDR_INTS) hdr[i] = 0;
}

// wT[t][o][i] = w[t][i][o]  (so B fragments become contiguous b64 loads)
__global__ void transpose_w_kernel(const float* __restrict__ w, float* __restrict__ wT) {
    int i = blockIdx.x * blockDim.x + threadIdx.x;   // over NT*D*D
    if (i >= NT * D * D) return;
    int t  = i / (D * D);
    int r  = i % (D * D);
    int ki = r / D;          // input index
    int o  = r % D;          // output index
    wT[(size_t)t * D * D + (size_t)o * D + ki] = w[i];
}

__global__ void hist_kernel(const int* __restrict__ et, int E, int* counts) {
    __shared__ int h[NT];
    if (threadIdx.x < NT) h[threadIdx.x] = 0;
    __syncthreads();
    int i = blockIdx.x * blockDim.x + threadIdx.x;
    if (i < E) atomicAdd(&h[et[i]], 1);
    __syncthreads();
    if (threadIdx.x < NT) atomicAdd(&counts[threadIdx.x], h[threadIdx.x]);
}

__global__ void prefix_kernel(int* hdr) {
    // single thread: 16-aligned exclusive prefix so every tile is one type
    int run = 0;
    for (int t = 0; t < NT; ++t) {
        hdr[8 + t]  = run;   // starts[t]
        hdr[17 + t] = run;   // cursors[t]
        run = (run + hdr[t] + TILE - 1) & ~(TILE - 1);
    }
    hdr[8 + NT] = run;       // total padded edge count
}

__global__ void scatter_kernel(const int* __restrict__ src, const int* __restrict__ dst,
                               const int* __restrict__ et, int E,
                               int* cursors, int* ssrc, int* sdst) {
    __shared__ int cnt[NT];
    __shared__ int base[NT];
    if (threadIdx.x < NT) cnt[threadIdx.x] = 0;
    __syncthreads();
    int i = blockIdx.x * blockDim.x + threadIdx.x;
    int t = 0, my = 0;
    bool act = (i < E);
    if (act) { t = et[i]; my = atomicAdd(&cnt[t], 1); }
    __syncthreads();
    if (threadIdx.x < NT)
        base[threadIdx.x] = atomicAdd(&cursors[threadIdx.x], cnt[threadIdx.x]);
    __syncthreads();
    if (act) { int p = base[t] + my; ssrc[p] = src[i]; sdst[p] = dst[i]; }
}

// ---------------- WMMA kernels (fp32, 16x16x4) ----------------

// out[r, :] = x[r, :] @ slw^T  (B[k][n] = slw[n*D + k]) ; initializes d_out
__global__ void __launch_bounds__(32)
selfloop_gemm_kernel(const float* __restrict__ x, const float* __restrict__ slw,
                     float* __restrict__ out, int NN) {
    const int lane = threadIdx.x;
    const int half = lane >> 4;        // 0: K pair {0,1}, 1: K pair {2,3}
    const int lq   = lane & 15;
    const int rowBase = blockIdx.x * TILE;
    // A-row m only affects D-row m; tail D-rows are not stored, so clamping is safe
    int m = rowBase + lq;
    if (m >= NN) m = NN - 1;
    const float* arow = x + (size_t)m * D + half * 2;

    v8f acc0 = {}, acc1 = {}, acc2 = {}, acc3 = {};
#pragma unroll
    for (int kb = 0; kb < 16; ++kb) {
        const int k = kb * 4;
        v2f a  = *(const v2f*)(arow + k);
        // slw is row-major [out][in] -> B[k][n] pairs are contiguous in k
        v2f b0 = *(const v2f*)(slw + (size_t)(lq +  0) * D + k + half * 2);
        v2f b1 = *(const v2f*)(slw + (size_t)(lq + 16) * D + k + half * 2);
        v2f b2 = *(const v2f*)(slw + (size_t)(lq + 32) * D + k + half * 2);
        v2f b3 = *(const v2f*)(slw + (size_t)(lq + 48) * D + k + half * 2);
        acc0 = __builtin_amdgcn_wmma_f32_16x16x4_f32(false, a, false, b0, (short)0, acc0, false, false);
        acc1 = __builtin_amdgcn_wmma_f32_16x16x4_f32(false, a, false, b1, (short)0, acc1, false, false);
        acc2 = __builtin_amdgcn_wmma_f32_16x16x4_f32(false, a, false, b2, (short)0, acc2, false, false);
        acc3 = __builtin_amdgcn_wmma_f32_16x16x4_f32(false, a, false, b3, (short)0, acc3, false, false);
    }
    // D layout: VGPR j -> row (half*8 + j), col lq (+16 per N tile)
#pragma unroll
    for (int j = 0; j < 8; ++j) {
        int r = rowBase + half * 8 + j;
        if (r < NN) {
            float* o = out + (size_t)r * D + lq;
            o[0]  = acc0[j];
            o[16] = acc1[j];
            o[32] = acc2[j];
            o[48] = acc3[j];
        }
    }
}

// per 16-edge single-type tile: out[dst] += x[src] @ W[type]
// wT is [type][out][in] so B fragments are contiguous b64 loads.
__global__ void __launch_bounds__(32)
edge_gemm_kernel(const float* __restrict__ x, const float* __restrict__ wT,
                 const int* __restrict__ hdr, const int* __restrict__ ssrc,
                 const int* __restrict__ sdst, float* out) {
    const int tileBase = blockIdx.x * TILE;
    const int total = hdr[8 + NT];
    if (tileBase >= total) return;                 // uniform per wave

    int t = 0;
#pragma unroll
    for (int q = 0; q < NT - 1; ++q)
        if (tileBase >= hdr[8 + q + 1]) t = q + 1;
    const float* wt = wT + (size_t)t * D * D;      // [out][in]

    const int lane = threadIdx.x;
    const int half = lane >> 4;
    const int lq   = lane & 15;
    // Padding rows (s==-1): clamp to row 0; their D-rows are discarded (dst==-1)
    int s = ssrc[tileBase + lq];
    int sr = (s < 0) ? 0 : s;
    const float* arow = x + (size_t)sr * D + half * 2;

    v8f acc0 = {}, acc1 = {}, acc2 = {}, acc3 = {};
#pragma unroll
    for (int kb = 0; kb < 16; ++kb) {
        const int k = kb * 4;
        v2f a  = *(const v2f*)(arow + k);
        v2f b0 = *(const v2f*)(wt + (size_t)(lq +  0) * D + k + half * 2);
        v2f b1 = *(const v2f*)(wt + (size_t)(lq + 16) * D + k + half * 2);
        v2f b2 = *(const v2f*)(wt + (size_t)(lq + 32) * D + k + half * 2);
        v2f b3 = *(const v2f*)(wt + (size_t)(lq + 48) * D + k + half * 2);
        acc0 = __builtin_amdgcn_wmma_f32_16x16x4_f32(false, a, false, b0, (short)0, acc0, false, false);
        acc1 = __builtin_amdgcn_wmma_f32_16x16x4_f32(false, a, false, b1, (short)0, acc1, false, false);
        acc2 = __builtin_amdgcn_wmma_f32_16x16x4_f32(false, a, false, b2, (short)0, acc2, false, false);
        acc3 = __builtin_amdgcn_wmma_f32_16x16x4_f32(false, a, false, b3, (short)0, acc3, false, false);
    }
#pragma unroll
    for (int j = 0; j < 8; ++j) {
        int d = sdst[tileBase + half * 8 + j];
        if (d >= 0) {
            float* o = out + (size_t)d * D + lq;
            atomicAdd(o + 0,  acc0[j]);
            atomicAdd(o + 16, acc1[j]);
            atomicAdd(o + 32, acc2[j]);
            atomicAdd(o + 48, acc3[j]);
        }
    }
}

// ---------------- fallback (ws too small): scalar per-edge GEMV ----------------

__global__ void naive_edge_kernel(const float* __restrict__ x, const int* __restrict__ src,
                                  const int* __restrict__ dst, const int* __restrict__ et,
                                  const float* __restrict__ w, float* out, int E) {
    int e = blockIdx.x * blockDim.x + threadIdx.x;
    if (e >= E) return;
    const float* xr = x + (size_t)src[e] * D;
    const float* wt = w + (size_t)et[e] * D * D;
    float* o = out + (size_t)dst[e] * D;
    for (int oo = 0; oo < D; ++oo) {
        float sum = 0.0f;
        for (int i = 0; i < D; ++i) sum += xr[i] * wt[(size_t)i * D + oo];
        atomicAdd(o + oo, sum);
    }
}

// ---------------- launch ----------------

extern "C" void kernel_launch(void* const* d_in, const int* in_sizes, int n_in,
                              void* d_out, int out_size, void* d_ws, size_t ws_size,
                              hipStream_t stream) {
    const float* x    = (const float*)d_in[0];
    const int*   eidx = (const int*)d_in[1];
    const int*   et   = (const int*)d_in[2];
    const float* w    = (const float*)d_in[3];
    const float* slw  = (const float*)d_in[4];
    const int NN = in_sizes[0] / D;
    const int E  = in_sizes[1] / 2;
    const int* src = eidx;
    const int* dst = eidx + E;
    float* out = (float*)d_out;

    // 1) self-loop GEMM initializes all of d_out
    selfloop_gemm_kernel<<<(NN + TILE - 1) / TILE, 32, 0, stream>>>(x, slw, out, NN);

    const int CAP = E + NT * TILE;   // room for per-type 16-padding
    // ws layout: hdr (64 ints) | wT (NT*D*D floats) | ssrc (CAP) | sdst (CAP)
    const size_t need = (size_t)(HDR_INTS + NT * D * D + 2 * CAP) * sizeof(int);
    if (ws_size >= need) {
        int*   hdr  = (int*)d_ws;
        float* wT   = (float*)(hdr + HDR_INTS);
        int*   ssrc = (int*)(wT + NT * D * D);
        int*   sdst = ssrc + CAP;
        init_ws_kernel<<<(CAP + 255) / 256, 256, 0, stream>>>(hdr, ssrc, CAP);
        transpose_w_kernel<<<(NT * D * D + 255) / 256, 256, 0, stream>>>(w, wT);
        hist_kernel<<<(E + 255) / 256, 256, 0, stream>>>(et, E, hdr);
        prefix_kernel<<<1, 1, 0, stream>>>(hdr);
        scatter_kernel<<<(E + 255) / 256, 256, 0, stream>>>(src, dst, et, E,
                                                            hdr + 17, ssrc, sdst);
        edge_gemm_kernel<<<(CAP + TILE - 1) / TILE, 32, 0, stream>>>(x, wT, hdr, ssrc, sdst, out);
    } else {
        naive_edge_kernel<<<(E + 255) / 256, 256, 0, stream>>>(x, src, dst, et, w, out, E);
    }
}